// MultiHeadAttentionDecoder_48095043781186
// MI455X (gfx1250) — compile-verified
//
#include <hip/hip_runtime.h>

// ---------------- problem constants ----------------
#define BB     4
#define TNEW   512
#define DMODEL 2048
#define NH     16
#define DH     128
#define TPAST  1536
#define TTOT   2048
#define SROW   2052   // LDS score row stride (floats), not a multiple of 64 banks

static constexpr size_t OFF_ATTN = (size_t)BB * TNEW * DMODEL;                 //  4,194,304
static constexpr size_t OFF_K    = OFF_ATTN + (size_t)BB * NH * TNEW * TTOT;   // 71,303,168
static constexpr size_t OFF_V    = OFF_K + (size_t)BB * NH * TTOT * DH;        // 88,080,384

typedef __attribute__((ext_vector_type(2))) float v2f;
typedef __attribute__((ext_vector_type(8))) float v8f;

// D = A(16x4) * B(4x16) + C   (f32 WMMA, wave32)
__device__ __forceinline__ v8f wmma_f32(v2f a, v2f b, v8f c) {
  return __builtin_amdgcn_wmma_f32_16x16x4_f32(false, a, false, b, (short)0, c,
                                               false, false);
}

// ---------------- K1: copy past_k/past_v into concatenated k/v outputs -------
__global__ void __launch_bounds__(256)
k_copy_past(const float4* __restrict__ pk, const float4* __restrict__ pv,
            float4* __restrict__ kout, float4* __restrict__ vout) {
  const int srcPerBH = TPAST * DH / 4;  // 49152 float4 per (b,h)
  const int dstPerBH = TTOT * DH / 4;   // 65536 float4 per (b,h)
  const int n = BB * NH * srcPerBH;     // 3,145,728
  int i = blockIdx.x * blockDim.x + threadIdx.x;
  if (i < n) {
    int bh = i / srcPerBH, r = i - bh * srcPerBH;
    kout[(size_t)bh * dstPerBH + r] = pk[i];
  } else {
    int j = i - n;
    if (j < n) {
      int bh = j / srcPerBH, r = j - bh * srcPerBH;
      vout[(size_t)bh * dstPerBH + r] = pv[j];
    }
  }
}

// ---------------- K2: Q/K/V projections (wave per 16x16 tile) ----------------
__global__ void __launch_bounds__(128)
k_qkv(const float* __restrict__ x,
      const float* __restrict__ Wq, const float* __restrict__ Wk,
      const float* __restrict__ Wv,
      float* __restrict__ qout, float* __restrict__ kout, float* __restrict__ vout) {
  int wave = (blockIdx.x * blockDim.x + threadIdx.x) >> 5;
  int lane = threadIdx.x & 31;
  // 64 bh * 32 row-tiles * 8 col-tiles = 16384 tiles
  int bh  = wave >> 8;
  int rem = wave & 255;
  int rt = rem >> 3, ct = rem & 7;
  int b = bh >> 4, h = bh & 15;

  const float* xA = x + (size_t)b * TNEW * DMODEL + (size_t)h * DH; // A[t][d] = xA[t*DMODEL+d]
  int lr = lane & 15;
  int g2 = (lane >> 4) * 2;
  int r  = rt * 16 + lr;
  int n  = ct * 16 + lr;

  // preload A fragments for K=128 (32 steps of K=4); MUST stay fully unrolled
  // everywhere it is indexed so afr[] lives in VGPRs (no scratch demotion).
  v2f afr[32];
#pragma unroll
  for (int s = 0; s < 32; ++s) {
    const float2 t2 = *(const float2*)(xA + (size_t)r * DMODEL + 4 * s + g2);
    afr[s].x = t2.x; afr[s].y = t2.y;
  }

  v8f cq = {}, ck = {}, cv = {};
#pragma unroll
  for (int s = 0; s < 32; ++s) {
    int kk = 4 * s + g2;
    // B[kk][n] = W[n][kk]  (torch Linear: y = x @ W^T)
    const float2 bq = *(const float2*)(Wq + (size_t)n * DH + kk);
    const float2 bk = *(const float2*)(Wk + (size_t)n * DH + kk);
    const float2 bv = *(const float2*)(Wv + (size_t)n * DH + kk);
    v2f vb;
    vb.x = bq.x; vb.y = bq.y; cq = wmma_f32(afr[s], vb, cq);
    vb.x = bk.x; vb.y = bk.y; ck = wmma_f32(afr[s], vb, ck);
    vb.x = bv.x; vb.y = bv.y; cv = wmma_f32(afr[s], vb, cv);
  }

  int rbase = rt * 16 + ((lane >> 4) << 3);
  size_t qbase  = (size_t)bh * TNEW * DH;             // q laid out (B,H,Tnew,Dh)
  size_t kvbase = ((size_t)bh * TTOT + TPAST) * DH;   // new rows at [Tpast..)
#pragma unroll
  for (int i = 0; i < 8; ++i) {
    int row = rbase + i;
    qout[qbase + (size_t)row * DH + n]  = cq[i];
    kout[kvbase + (size_t)row * DH + n] = ck[i];
    vout[kvbase + (size_t)row * DH + n] = cv[i];
  }
}

// ---------------- K3: fused scores -> softmax -> P@V -------------------------
// One 128-thread (4-wave) WG per (b,h,16-query tile). Scores in 128KB LDS.
__global__ void __launch_bounds__(128)
k_attn(const float* __restrict__ q, const float* __restrict__ kall,
       const float* __restrict__ vall, const unsigned char* __restrict__ pad,
       float* __restrict__ attn, float* __restrict__ oh) {
  extern __shared__ float sc[];          // [16][SROW]
  __shared__ float red[16 * 8];

  int wg = blockIdx.x;                   // 64 bh * 32 qtiles = 2048
  int bh = wg >> 5, qt = wg & 31;
  int b  = bh >> 4;
  int tid = threadIdx.x;
  int wave = tid >> 5, lane = tid & 31;
  int lr = lane & 15;
  int g2 = (lane >> 4) * 2;
  const float scale = 0.0883883476483184f;   // 1/sqrt(128)

  const float* qbase = q + ((size_t)bh * TNEW + (size_t)qt * 16) * DH;
  const float* kbase = kall + (size_t)bh * TTOT * DH;

  // preload the 16x128 q tile as 32 A fragments (reused across all key tiles);
  // fully unrolled indexing keeps them in VGPRs.
  v2f afr[32];
#pragma unroll
  for (int s = 0; s < 32; ++s) {
    const float2 t2 = *(const float2*)(qbase + (size_t)lr * DH + 4 * s + g2);
    afr[s].x = t2.x; afr[s].y = t2.y;
  }

  // ---- step 1: scores tile (16 x 2048) into LDS, masked ----
  for (int ct = wave; ct < TTOT / 16; ct += 4) {
    int n = ct * 16 + lr;                // key index
    v8f c = {};
#pragma unroll
    for (int s = 0; s < 32; ++s) {
      // B[kk][n] = k[n][kk]  -> contiguous float2 along d
      const float2 t2 = *(const float2*)(kbase + (size_t)n * DH + 4 * s + g2);
      v2f vb; vb.x = t2.x; vb.y = t2.y;
      c = wmma_f32(afr[s], vb, c);
    }
    int rb = (lane >> 4) << 3;
    bool padok = pad[(size_t)b * TTOT + n] != 0;
#pragma unroll
    for (int i = 0; i < 8; ++i) {
      int row = rb + i;                  // local query row 0..15
      int tglob = qt * 16 + row;
      float val = c[i] * scale;
      if (!padok || n > TPAST + tglob) val = -__builtin_inff();
      sc[row * SROW + n] = val;
    }
  }
  __syncthreads();

  // ---- step 2: row softmax (16 rows x 8 threads each), write attn once ----
  {
    int row = tid >> 3, sub = tid & 7;
    float m = -__builtin_inff();
    for (int i = 0; i < TTOT / 8; ++i)
      m = fmaxf(m, sc[row * SROW + sub + 8 * i]);
    red[row * 8 + sub] = m;
    __syncthreads();
    float rm = red[row * 8];
#pragma unroll
    for (int i = 1; i < 8; ++i) rm = fmaxf(rm, red[row * 8 + i]);
    __syncthreads();
    float ssum = 0.f;
    for (int i = 0; i < TTOT / 8; ++i) {
      int j = sub + 8 * i;
      float p = __expf(sc[row * SROW + j] - rm);
      sc[row * SROW + j] = p;
      ssum += p;
    }
    red[row * 8 + sub] = ssum;
    __syncthreads();
    float tot = 0.f;
#pragma unroll
    for (int i = 0; i < 8; ++i) tot += red[row * 8 + i];
    float inv = 1.0f / tot;
    size_t abase = ((size_t)bh * TNEW + (size_t)qt * 16 + row) * TTOT;
    for (int i = 0; i < TTOT / 8; ++i) {
      int j = sub + 8 * i;
      float p = sc[row * SROW + j] * inv;
      sc[row * SROW + j] = p;
      attn[abase + j] = p;
    }
  }
  __syncthreads();

  // ---- step 3: O(16x128) = P(16x2048) @ V(2048x128), A-frags from LDS ----
  const float* vbase = vall + (size_t)bh * TTOT * DH;
  v8f c0 = {}, c1 = {};
  int n0 = wave * 16 + lr, n1 = (wave + 4) * 16 + lr;
#pragma unroll 4
  for (int s = 0; s < TTOT / 4; ++s) {
    int kk = 4 * s + g2;
    const float2 t2 = *(const float2*)(&sc[lr * SROW + kk]);
    v2f a; a.x = t2.x; a.y = t2.y;
    v2f b0, b1;
    b0.x = vbase[(size_t)kk * DH + n0]; b0.y = vbase[(size_t)(kk + 1) * DH + n0];
    b1.x = vbase[(size_t)kk * DH + n1]; b1.y = vbase[(size_t)(kk + 1) * DH + n1];
    c0 = wmma_f32(a, b0, c0);
    c1 = wmma_f32(a, b1, c1);
  }
  int rb = (lane >> 4) << 3;
  size_t obase = ((size_t)bh * TNEW + (size_t)qt * 16) * DH;
#pragma unroll
  for (int i = 0; i < 8; ++i) {
    int row = rb + i;
    oh[obase + (size_t)row * DH + n0] = c0[i];
    oh[obase + (size_t)row * DH + n1] = c1[i];
  }
}

// ---------------- K4: out = O(2048x2048) @ Wout^T + b_out --------------------
__global__ void __launch_bounds__(128)
k_outproj(const float* __restrict__ oh, const float* __restrict__ Wout,
          const float* __restrict__ bias, float* __restrict__ out) {
  int wave = (blockIdx.x * blockDim.x + threadIdx.x) >> 5;
  int lane = threadIdx.x & 31;
  int rt = wave >> 7, ct = wave & 127;   // 128 x 128 tiles
  int lr = lane & 15;
  int g2 = (lane >> 4) * 2;
  int grow = rt * 16 + lr;               // global row = b*512 + t
  int bb = grow >> 9, t = grow & 511;
  int n = ct * 16 + lr;                  // output column

  v8f c = {};
#pragma unroll 4
  for (int s = 0; s < DMODEL / 4; ++s) {
    int kk = 4 * s + g2;                 // even -> (h,e) and (h,e+1) same head
    int h = kk >> 7, e = kk & 127;
    const float2 a2 =
        *(const float2*)(oh + (((size_t)(bb * NH + h) * TNEW + t) * DH + e));
    const float2 b2 = *(const float2*)(Wout + (size_t)n * DMODEL + kk);
    v2f a; a.x = a2.x; a.y = a2.y;
    v2f b; b.x = b2.x; b.y = b2.y;
    c = wmma_f32(a, b, c);
  }
  float bv = bias[n];
  int rb = rt * 16 + ((lane >> 4) << 3);
#pragma unroll
  for (int i = 0; i < 8; ++i) {
    int row = rb + i;
    out[(size_t)row * DMODEL + n] = c[i] + bv;
  }
}

// ---------------- launcher ---------------------------------------------------
extern "C" void kernel_launch(void* const* d_in, const int* in_sizes, int n_in,
                              void* d_out, int out_size, void* d_ws, size_t ws_size,
                              hipStream_t stream) {
  const float* x            = (const float*)d_in[0];
  const unsigned char* pad  = (const unsigned char*)d_in[1]; // bool mask
  const float* past_k       = (const float*)d_in[2];
  const float* past_v       = (const float*)d_in[3];
  const float* Wq           = (const float*)d_in[4];
  const float* Wk           = (const float*)d_in[5];
  const float* Wv           = (const float*)d_in[6];
  const float* Wout         = (const float*)d_in[7];
  const float* b_out        = (const float*)d_in[8];

  float* out  = (float*)d_out;
  float* attn = out + OFF_ATTN;
  float* kout = out + OFF_K;
  float* vout = out + OFF_V;

  float* q  = (float*)d_ws;                              // 16.8 MB
  float* oh = q + (size_t)BB * NH * TNEW * DH;           // 16.8 MB

  // 1) concat copies of past K/V
  k_copy_past<<<24576, 256, 0, stream>>>((const float4*)past_k,
                                         (const float4*)past_v,
                                         (float4*)kout, (float4*)vout);
  // 2) QKV projections
  k_qkv<<<4096, 128, 0, stream>>>(x, Wq, Wk, Wv, q, kout, vout);
  // 3) fused scores + softmax + P@V  (128KB+ dynamic LDS per WG)
  k_attn<<<2048, 128, (size_t)16 * SROW * sizeof(float), stream>>>(
      q, kout, vout, pad, attn, oh);
  // 4) output projection + bias
  k_outproj<<<4096, 128, 0, stream>>>(oh, Wout, b_out, out);
}